// HMM_10385230922018
// MI455X (gfx1250) — compile-verified
//
#include <hip/hip_runtime.h>
#include <stdint.h>

#ifndef __has_builtin
#define __has_builtin(x) 0
#endif

#define LN2f   0.69314718055994530942f
#define LOG2Ef 1.44269504088896340736f
#define NEGC   (-100.0f)

// workspace layout (floats)
#define OFF_EXPA   0      // 16 floats: expA[i*4+j] = exp(log_A[i][j])
#define OFF_LOGPI  16     // 4 floats
#define OFF_LOGBT  32     // 8000*4 floats: logB_T[v*4+s]
#define OFF_BT     32032  // 8000*4 floats: B_T[v*4+s] = exp(logB_T)

// ---------------------------------------------------------------------------
// Kernel 1: parameter preprocessing (log_softmax + masks), builds tables in ws
// ---------------------------------------------------------------------------
__global__ void hmm_prep_kernel(const float* __restrict__ init_pi,
                                const float* __restrict__ init_A,
                                const float* __restrict__ init_B,
                                float* __restrict__ ws) {
  __shared__ float red[256];
  const int tid = threadIdx.x;

  for (int s = 0; s < 4; ++s) {
    const float* row = init_B + s * 8000;
    float mx = -3.0e38f;
    for (int v = tid; v < 8000; v += 256) mx = fmaxf(mx, row[v]);
    red[tid] = mx; __syncthreads();
    for (int off = 128; off > 0; off >>= 1) {
      if (tid < off) red[tid] = fmaxf(red[tid], red[tid + off]);
      __syncthreads();
    }
    mx = red[0]; __syncthreads();

    float sum = 0.0f;
    for (int v = tid; v < 8000; v += 256)
      sum += __builtin_amdgcn_exp2f((row[v] - mx) * LOG2Ef);
    red[tid] = sum; __syncthreads();
    for (int off = 128; off > 0; off >>= 1) {
      if (tid < off) red[tid] += red[tid + off];
      __syncthreads();
    }
    float lse = mx + __builtin_amdgcn_logf(red[0]) * LN2f;
    __syncthreads();

    for (int v = tid; v < 8000; v += 256) {
      float lb = fmaxf(row[v] - lse, NEGC);
      ws[OFF_LOGBT + v * 4 + s] = lb;
      ws[OFF_BT    + v * 4 + s] = __builtin_amdgcn_exp2f(lb * LOG2Ef);
    }
    __syncthreads();
  }

  if (tid == 0) {
    // log_pi with mask [0, NEG, NEG, 0]
    float x0 = init_pi[0], x1 = init_pi[1] + NEGC, x2 = init_pi[2] + NEGC, x3 = init_pi[3];
    float mx = fmaxf(fmaxf(x0, x1), fmaxf(x2, x3));
    float sum = __builtin_amdgcn_exp2f((x0 - mx) * LOG2Ef) +
                __builtin_amdgcn_exp2f((x1 - mx) * LOG2Ef) +
                __builtin_amdgcn_exp2f((x2 - mx) * LOG2Ef) +
                __builtin_amdgcn_exp2f((x3 - mx) * LOG2Ef);
    float lse = mx + __builtin_amdgcn_logf(sum) * LN2f;
    ws[OFF_LOGPI + 0] = fmaxf(x0 - lse, NEGC);
    ws[OFF_LOGPI + 1] = fmaxf(x1 - lse, NEGC);
    ws[OFF_LOGPI + 2] = fmaxf(x2 - lse, NEGC);
    ws[OFF_LOGPI + 3] = fmaxf(x3 - lse, NEGC);

    // expA = exp(log_softmax(init_A + mask)) rows; pattern rows of allowed transitions
    const int pat[16] = {0,1,1,0, 0,1,1,0, 1,0,0,1, 1,0,0,1};
    for (int i = 0; i < 4; ++i) {
      float xa[4];
      for (int j = 0; j < 4; ++j)
        xa[j] = init_A[i * 4 + j] + (pat[i * 4 + j] ? 0.0f : NEGC);
      float m2 = fmaxf(fmaxf(xa[0], xa[1]), fmaxf(xa[2], xa[3]));
      float s2 = 0.0f;
      for (int j = 0; j < 4; ++j) s2 += __builtin_amdgcn_exp2f((xa[j] - m2) * LOG2Ef);
      float l2 = m2 + __builtin_amdgcn_logf(s2) * LN2f;
      for (int j = 0; j < 4; ++j)
        ws[OFF_EXPA + i * 4 + j] =
            __builtin_amdgcn_exp2f(fmaxf(xa[j] - l2, NEGC) * LOG2Ef);
    }
  }
}

// ---------------------------------------------------------------------------
// Kernel 2: forward scan. 1 thread = 1 sequence. B table staged into LDS by TDM.
// ---------------------------------------------------------------------------
#if __has_builtin(__builtin_amdgcn_tensor_load_to_lds)
#define HAVE_TDM 1
typedef unsigned int uint32x4 __attribute__((ext_vector_type(4)));
typedef int          int32x4  __attribute__((ext_vector_type(4)));
typedef int          int32x8  __attribute__((ext_vector_type(8)));
#endif

// One recurrence step. Uses/updates: a0..a3 (scaled probs), c (log offset),
// p0..p3 (previous log-alphas), writes component COMP of ob0..ob3.
#define STEP(yv, mv, COMP) do {                                               \
    float4 _b = ldsB[(yv)];                                                   \
    float _v0 = fmaf(a0, eA[0],  fmaf(a1, eA[4],  fmaf(a2, eA[8],  a3 * eA[12]))) * _b.x; \
    float _v1 = fmaf(a0, eA[1],  fmaf(a1, eA[5],  fmaf(a2, eA[9],  a3 * eA[13]))) * _b.y; \
    float _v2 = fmaf(a0, eA[2],  fmaf(a1, eA[6],  fmaf(a2, eA[10], a3 * eA[14]))) * _b.z; \
    float _v3 = fmaf(a0, eA[3],  fmaf(a1, eA[7],  fmaf(a2, eA[11], a3 * eA[15]))) * _b.w; \
    float _l0 = __builtin_amdgcn_logf(_v0);                                   \
    float _l1 = __builtin_amdgcn_logf(_v1);                                   \
    float _l2 = __builtin_amdgcn_logf(_v2);                                   \
    float _l3 = __builtin_amdgcn_logf(_v3);                                   \
    float _o0 = fmaf(_l0, LN2f, c), _o1 = fmaf(_l1, LN2f, c);                 \
    float _o2 = fmaf(_l2, LN2f, c), _o3 = fmaf(_l3, LN2f, c);                 \
    float _m = (mv);                                                          \
    if (_m == 1.0f) {                                                         \
      float _lm = fmaxf(fmaxf(_l0, _l1), fmaxf(_l2, _l3));                    \
      float _vm = fmaxf(fmaxf(_v0, _v1), fmaxf(_v2, _v3));                    \
      float _r  = __builtin_amdgcn_rcpf(_vm);                                 \
      a0 = _v0 * _r; a1 = _v1 * _r; a2 = _v2 * _r; a3 = _v3 * _r;             \
      c = fmaf(_lm, LN2f, c);                                                 \
    } else if (_m == 0.0f) {                                                  \
      _o0 = p0; _o1 = p1; _o2 = p2; _o3 = p3;                                 \
    } else {                                                                  \
      _o0 = _m * _o0 + (1.0f - _m) * p0;                                      \
      _o1 = _m * _o1 + (1.0f - _m) * p1;                                      \
      _o2 = _m * _o2 + (1.0f - _m) * p2;                                      \
      _o3 = _m * _o3 + (1.0f - _m) * p3;                                      \
      c = fmaxf(fmaxf(_o0, _o1), fmaxf(_o2, _o3));                            \
      a0 = __builtin_amdgcn_exp2f((_o0 - c) * LOG2Ef);                        \
      a1 = __builtin_amdgcn_exp2f((_o1 - c) * LOG2Ef);                        \
      a2 = __builtin_amdgcn_exp2f((_o2 - c) * LOG2Ef);                        \
      a3 = __builtin_amdgcn_exp2f((_o3 - c) * LOG2Ef);                        \
    }                                                                         \
    p0 = _o0; p1 = _o1; p2 = _o2; p3 = _o3;                                   \
    ob0.COMP = _o0; ob1.COMP = _o1; ob2.COMP = _o2; ob3.COMP = _o3;           \
  } while (0)

__global__ void hmm_forward_kernel(const int* __restrict__ Y,
                                   const float* __restrict__ mask,
                                   const float* __restrict__ ws,
                                   float* __restrict__ out, int N) {
  extern __shared__ float4 smem[];           // 8000 x float4 = 128000 bytes
  const float4* ldsB = smem;

  // ---- stage B_T (linear probs) into LDS --------------------------------
#if HAVE_TDM
  if (threadIdx.x == 0) {
    uint64_t ga = (uint64_t)(uintptr_t)(ws + OFF_BT);
    unsigned ldsOff = (unsigned)(uintptr_t)(void*)smem;   // LDS aperture: low 32 bits = offset
    uint32x4 g0 = { 1u,                                   // count=1 (valid), no gather
                    ldsOff,                               // lds_addr
                    (unsigned)(ga & 0xffffffffu),         // global_addr[31:0]
                    (unsigned)(((ga >> 32) & 0x01ffffffu) | (2u << 30)) }; // [56:32] | type=2
    int32x8 g1 = { (int)(2u << 16),          // data_size = 4B
                   (int)(0x7D00u << 16),     // tensor_dim0 = 32000 (low 16)
                   (int)(1u << 16),          // tensor_dim1 = 1
                   (int)(0x7D00u << 16),     // tile_dim0 = 32000
                   1,                        // tile_dim1 = 1
                   32000,                    // tensor_dim0_stride[31:0]
                   0, 0 };
    int32x4 z4 = { 0, 0, 0, 0 };
#if __clang_major__ >= 23
    int32x8 z8 = { 0, 0, 0, 0, 0, 0, 0, 0 };
    __builtin_amdgcn_tensor_load_to_lds(g0, g1, z4, z4, z8, 0);
#else
    __builtin_amdgcn_tensor_load_to_lds(g0, g1, z4, z4, 0);
#endif
    __builtin_amdgcn_s_wait_tensorcnt(0);
  }
  __syncthreads();
#else
  {
    const float4* gBT = (const float4*)(ws + OFF_BT);
    for (int i = threadIdx.x; i < 8000; i += blockDim.x) smem[i] = gBT[i];
    __syncthreads();
  }
#endif

  const int n = blockIdx.x * blockDim.x + threadIdx.x;
  if (n >= N) return;

  // uniform parameters (scalar loads)
  float eA[16];
#pragma unroll
  for (int k = 0; k < 16; ++k) eA[k] = ws[OFF_EXPA + k];
  float lp0 = ws[OFF_LOGPI + 0], lp1 = ws[OFF_LOGPI + 1];
  float lp2 = ws[OFF_LOGPI + 2], lp3 = ws[OFF_LOGPI + 3];

  const int*   Yrow = Y    + (size_t)n * 1024;
  const float* Mrow = mask + (size_t)n * 1024;
  float*       outp = out  + (size_t)n * 4096;   // out[n][s][t], s-stride = 1024
  const float* gLB  = ws + OFF_LOGBT;

  int4   ych = *(const int4*)Yrow;
  float4 mch = *(const float4*)Mrow;

  // ---- t = 0 in log space (exact NEG clamps) ----------------------------
  int y0 = ych.x;
  float p0 = lp0 + gLB[y0 * 4 + 0];
  float p1 = lp1 + gLB[y0 * 4 + 1];
  float p2 = lp2 + gLB[y0 * 4 + 2];
  float p3 = lp3 + gLB[y0 * 4 + 3];
  float c  = fmaxf(fmaxf(p0, p1), fmaxf(p2, p3));
  float a0 = __builtin_amdgcn_exp2f((p0 - c) * LOG2Ef);
  float a1 = __builtin_amdgcn_exp2f((p1 - c) * LOG2Ef);
  float a2 = __builtin_amdgcn_exp2f((p2 - c) * LOG2Ef);
  float a3 = __builtin_amdgcn_exp2f((p3 - c) * LOG2Ef);

  float4 ob0, ob1, ob2, ob3;
  ob0.x = p0; ob1.x = p1; ob2.x = p2; ob3.x = p3;

  STEP(ych.y, mch.y, y);
  STEP(ych.z, mch.z, z);
  STEP(ych.w, mch.w, w);
  *(float4*)(outp + 0)    = ob0;
  *(float4*)(outp + 1024) = ob1;
  *(float4*)(outp + 2048) = ob2;
  *(float4*)(outp + 3072) = ob3;

  for (int tb = 4; tb < 1024; tb += 4) {
    ych = *(const int4*)(Yrow + tb);
    mch = *(const float4*)(Mrow + tb);
    STEP(ych.x, mch.x, x);
    STEP(ych.y, mch.y, y);
    STEP(ych.z, mch.z, z);
    STEP(ych.w, mch.w, w);
    *(float4*)(outp + tb)        = ob0;
    *(float4*)(outp + 1024 + tb) = ob1;
    *(float4*)(outp + 2048 + tb) = ob2;
    *(float4*)(outp + 3072 + tb) = ob3;
  }
}

// ---------------------------------------------------------------------------
extern "C" void kernel_launch(void* const* d_in, const int* in_sizes, int n_in,
                              void* d_out, int out_size, void* d_ws, size_t ws_size,
                              hipStream_t stream) {
  const int*   Y       = (const int*)d_in[0];
  const float* mask    = (const float*)d_in[1];
  const float* init_pi = (const float*)d_in[2];
  const float* init_A  = (const float*)d_in[3];
  const float* init_B  = (const float*)d_in[4];
  float*       out     = (float*)d_out;
  float*       ws      = (float*)d_ws;

  const int T = 1024;
  const int N = in_sizes[0] / T;

  hipLaunchKernelGGL(hmm_prep_kernel, dim3(1), dim3(256), 0, stream,
                     init_pi, init_A, init_B, ws);

  const size_t shmem = 8000 * sizeof(float4);  // 128000 B (<= 320 KB/WGP)
  (void)hipFuncSetAttribute((const void*)hmm_forward_kernel,
                            hipFuncAttributeMaxDynamicSharedMemorySize,
                            (int)shmem);
  const int block = 64;                        // 2 wave32 waves
  hipLaunchKernelGGL(hmm_forward_kernel, dim3((N + block - 1) / block),
                     dim3(block), shmem, stream, Y, mask, ws, out, N);
}